// SAEncoder_31344671326224
// MI455X (gfx1250) — compile-verified
//
#include <hip/hip_runtime.h>

// ---------------- problem constants ----------------
#define B_    8
#define C_    96
#define C2_   192
#define H_    56
#define W_    56
#define N_    (H_*W_)          // 3136
#define M1_   (B_*N_)          // 25088 rows (b,n)
#define KNN_  9
#define ME_   (M1_*KNN_)       // 225792 edge rows
#define OC_   192
#define OHW_  (28*28)          // 784
#define EPS_  1e-5f

typedef __attribute__((ext_vector_type(2))) float v2f;
typedef __attribute__((ext_vector_type(8))) float v8f;

static __device__ __forceinline__ v8f wmma4(v2f a, v2f b, v8f c) {
  // V_WMMA_F32_16X16X4_F32 : D(16x16 f32) = A(16x4) * B(4x16) + C
  return __builtin_amdgcn_wmma_f32_16x16x4_f32(false, a, false, b, (short)0, c, false, false);
}

// ================= fc1: y[b,n,d] = sum_c x[b,c,n] W1[c,d] + b1[d] =================
// A fragments (stride-N scalar loads) kept in registers; wave sweeps all 6 col tiles.
// Full 96x96 W1 resident in LDS (36 KB).
__global__ void __launch_bounds__(256) fc1_gemm(const float* __restrict__ x,
                                                const float* __restrict__ Wf,
                                                const float* __restrict__ bias,
                                                float* __restrict__ y) {
  __shared__ float Wl[C_*C_];
  for (int t = threadIdx.x; t < C_*C_; t += 256) Wl[t] = Wf[t];

  int wave = threadIdx.x >> 5, lane = threadIdx.x & 31;
  int hi = lane >> 4, lp = lane & 15;
  int rowTile = (blockIdx.x * 8 + wave) * 16;
  int r = rowTile + lp;
  int b = r / N_, n = r % N_;
  const float* xrow = x + (size_t)b * C_ * N_ + n;   // channel stride = N_

  v2f afr[C_/4];                                     // 48 VGPRs, reused over col tiles
#pragma unroll
  for (int k = 0; k < C_/4; ++k) {
    int ka = 4*k + 2*hi;
    afr[k].x = xrow[(size_t)ka * N_];
    afr[k].y = xrow[(size_t)(ka+1) * N_];
  }
  __syncthreads();

  for (int ct = 0; ct < C_/16; ++ct) {
    int colBase = ct * 16;
    v8f acc = {};
#pragma unroll
    for (int k = 0; k < C_/4; ++k) {
      int ka = 4*k + 2*hi;
      v2f bb; bb.x = Wl[ka*C_ + colBase + lp]; bb.y = Wl[(ka+1)*C_ + colBase + lp];
      acc = wmma4(afr[k], bb, acc);
    }
    float bv = bias[colBase + lp];
#pragma unroll
    for (int i = 0; i < 8; ++i)
      y[(size_t)(rowTile + i + 8*hi) * C_ + colBase + lp] = acc[i] + bv;
  }
}

// ================= generic BN helpers =================
// channel-last layout: in[r*ch + c], one block per channel
__global__ void __launch_bounds__(256) stats_chlast(const float* __restrict__ in,
                                                    int rows, int ch,
                                                    float* __restrict__ sums) {
  int c = blockIdx.x;
  float s1 = 0.f, s2 = 0.f;
  for (int r = threadIdx.x; r < rows; r += 256) {
    float v = in[(size_t)r * ch + c];
    s1 += v; s2 += v * v;
  }
  __shared__ float red[512];
  red[threadIdx.x] = s1; red[256 + threadIdx.x] = s2;
  __syncthreads();
  for (int off = 128; off > 0; off >>= 1) {
    if ((int)threadIdx.x < off) {
      red[threadIdx.x]       += red[threadIdx.x + off];
      red[256 + threadIdx.x] += red[256 + threadIdx.x + off];
    }
    __syncthreads();
  }
  if (threadIdx.x == 0) { sums[c] = red[0]; sums[ch + c] = red[256]; }
}

// channel-middle layout: in[(b*ch + c)*n + j], one block per channel
__global__ void __launch_bounds__(256) stats_chmid(const float* __restrict__ in,
                                                   int ch, int n, int nb,
                                                   float* __restrict__ sums) {
  int c = blockIdx.x;
  float s1 = 0.f, s2 = 0.f;
  int tot = nb * n;
  for (int t = threadIdx.x; t < tot; t += 256) {
    int b = t / n, j = t - b * n;
    float v = in[((size_t)b * ch + c) * n + j];
    s1 += v; s2 += v * v;
  }
  __shared__ float red[512];
  red[threadIdx.x] = s1; red[256 + threadIdx.x] = s2;
  __syncthreads();
  for (int off = 128; off > 0; off >>= 1) {
    if ((int)threadIdx.x < off) {
      red[threadIdx.x]       += red[threadIdx.x + off];
      red[256 + threadIdx.x] += red[256 + threadIdx.x + off];
    }
    __syncthreads();
  }
  if (threadIdx.x == 0) { sums[c] = red[0]; sums[ch + c] = red[256]; }
}

__global__ void bn_finalize(const float* __restrict__ sums,
                            const float* __restrict__ gamma,
                            const float* __restrict__ beta,
                            int ch, float invCount, float* __restrict__ scsh) {
  int c = threadIdx.x;
  if (c < ch) {
    float mean = sums[c] * invCount;
    float var  = sums[ch + c] * invCount - mean * mean;
    float s = gamma[c] * rsqrtf(var + EPS_);
    scsh[c]      = s;
    scsh[ch + c] = beta[c] - mean * s;
  }
}

__global__ void bn_apply_chlast(const float* __restrict__ in,
                                const float* __restrict__ scsh,
                                int ch, size_t total, float* __restrict__ out) {
  size_t i = (size_t)blockIdx.x * 256 + threadIdx.x;
  if (i < total) {
    int c = (int)(i % ch);
    out[i] = in[i] * scsh[c] + scsh[ch + c];
  }
}

__global__ void bn_relu_chmid(const float* __restrict__ in,
                              const float* __restrict__ scsh,
                              int ch, int n, size_t total, float* __restrict__ out) {
  size_t i = (size_t)blockIdx.x * 256 + threadIdx.x;
  if (i < total) {
    int c = (int)((i / n) % ch);
    out[i] = fmaxf(in[i] * scsh[c] + scsh[ch + c], 0.f);
  }
}

__global__ void zero_f(float* p, size_t n) {
  size_t i = (size_t)blockIdx.x * 256 + threadIdx.x;
  if (i < n) p[i] = 0.f;
}

// ================= per-row squared norm of ft =================
__global__ void rowsq(const float* __restrict__ ft, float* __restrict__ sq) {
  int r = blockIdx.x * 256 + threadIdx.x;
  if (r < M1_) {
    const float4* p = (const float4*)(ft + (size_t)r * C_);
    float s = 0.f;
#pragma unroll
    for (int i = 0; i < C_/4; ++i) {
      float4 v = p[i];
      s += v.x*v.x + v.y*v.y + v.z*v.z + v.w*v.w;
    }
    sq[r] = s;
  }
}

// ================= kNN: WMMA Gram tiles + register top-9 =================
__global__ void __launch_bounds__(256) knn_topk(const float* __restrict__ ft,
                                                const float* __restrict__ sq,
                                                int* __restrict__ idxK) {
  __shared__ float tile[8][16][17];
  int wave = threadIdx.x >> 5, lane = threadIdx.x & 31;
  int hi = lane >> 4, lp = lane & 15;
  int gid = blockIdx.x * 8 + wave;
  int b  = gid / (N_/16);
  int qt = gid % (N_/16);
  int q0 = qt * 16;
  const float* ftb = ft + (size_t)b * N_ * C_;
  const float* sqb = sq + (size_t)b * N_;

  // A fragments for this 16-query block: resident in registers (48 VGPRs)
  v2f afr[C_/4];
  {
    const float* arow = ftb + (size_t)(q0 + lp) * C_;
#pragma unroll
    for (int k = 0; k < C_/4; ++k) {
      int ka = 4*k + 2*hi;
      afr[k].x = arow[ka]; afr[k].y = arow[ka+1];
    }
  }
  float sqq[8];
#pragma unroll
  for (int i = 0; i < 8; ++i) sqq[i] = sqb[q0 + i + 8*hi];

  float kd[KNN_]; int ki[KNN_];
#pragma unroll
  for (int j = 0; j < KNN_; ++j) { kd[j] = 3.4e38f; ki[j] = 0; }

  for (int mt = 0; mt < N_/16; ++mt) {
    int m0 = mt * 16;
    const float* brow = ftb + (size_t)(m0 + lp) * C_;
    float sqm = sqb[m0 + lp];
    v8f acc = {};
#pragma unroll
    for (int k = 0; k < C_/4; ++k) {
      int ka = 4*k + 2*hi;
      v2f bb; bb.x = brow[ka]; bb.y = brow[ka+1];
      acc = wmma4(afr[k], bb, acc);
    }
    // dist = |q|^2 + |m|^2 - 2<q,m>  -> stage tile in LDS for per-row scan
#pragma unroll
    for (int i = 0; i < 8; ++i)
      tile[wave][i + 8*hi][lp] = sqq[i] + sqm - 2.f * acc[i];
    __builtin_amdgcn_wave_barrier();
    asm volatile("s_wait_dscnt 0x0" ::: "memory");
    if (hi == 0) {
#pragma unroll
      for (int c = 0; c < 16; ++c) {
        float d = tile[wave][lp][c];
        if (d < kd[KNN_-1]) {
          kd[KNN_-1] = d; ki[KNN_-1] = b * N_ + m0 + c;   // store global ft row
#pragma unroll
          for (int j = KNN_-1; j > 0; --j) {
            if (kd[j] < kd[j-1]) {
              float td = kd[j]; kd[j] = kd[j-1]; kd[j-1] = td;
              int   ti = ki[j]; ki[j] = ki[j-1]; ki[j-1] = ti;
            }
          }
        }
      }
    }
    __builtin_amdgcn_wave_barrier();
    asm volatile("s_wait_dscnt 0x0" ::: "memory");
  }
  if (hi == 0) {
    size_t r = (size_t)b * N_ + q0 + lp;
#pragma unroll
    for (int j = 0; j < KNN_; ++j) idxK[r * KNN_ + j] = ki[j];
  }
}

// ================= edge MLP: [x_i, x_j - x_i] * W_edge (+bias) =================
// Gathered A fragments built ONCE per wave (96 VGPRs), reused across all 12
// column tiles; each 16-col W tile staged through LDS per iteration.
// STATS pass: accumulate per-channel sum/sumsq.  APPLY pass: BN+ReLU+max-over-K.
template <bool STATS>
__global__ void __launch_bounds__(256) edge_gemm(const float* __restrict__ ft,
                                                 const int* __restrict__ idxK,
                                                 const float* __restrict__ We,
                                                 const float* __restrict__ be,
                                                 const float* __restrict__ scsh,
                                                 float* __restrict__ gsum,
                                                 unsigned* __restrict__ gmax) {
  __shared__ float Wl[C2_*16];                        // 12 KB staging tile

  int wave = threadIdx.x >> 5, lane = threadIdx.x & 31;
  int hi = lane >> 4, lp = lane & 15;
  int rowTile = (blockIdx.x * 8 + wave) * 16;
  int rowM = rowTile + lp;
  int bn = rowM / KNN_;
  int kk = rowM - bn * KNN_;
  const float* fti = ft + (size_t)bn * C_;
  const float* ftj = ft + (size_t)idxK[(size_t)bn * KNN_ + kk] * C_;

  // Build A fragments once: channels 0..95 = x_i, 96..191 = x_j - x_i
  v2f afr[C2_/4];                                     // 96 VGPRs
#pragma unroll
  for (int k = 0; k < 24; ++k) {
    int ka = 4*k + 2*hi;
    afr[k].x = fti[ka]; afr[k].y = fti[ka+1];
  }
#pragma unroll
  for (int k = 24; k < 48; ++k) {
    int cc = 4*k + 2*hi - C_;
    afr[k].x = ftj[cc] - fti[cc]; afr[k].y = ftj[cc+1] - fti[cc+1];
  }

  for (int ct = 0; ct < C2_/16; ++ct) {
    int colBase = ct * 16;
    __syncthreads();                                  // prior tile reads done
    for (int t = threadIdx.x; t < C2_*16; t += 256)
      Wl[t] = We[(t >> 4) * C2_ + colBase + (t & 15)];
    __syncthreads();

    v8f acc = {};
#pragma unroll
    for (int k = 0; k < C2_/4; ++k) {
      int ka = 4*k + 2*hi;
      v2f bb; bb.x = Wl[ka*16 + lp]; bb.y = Wl[(ka+1)*16 + lp];
      acc = wmma4(afr[k], bb, acc);
    }

    int col = colBase + lp;
    float bv = be[col];
    if (STATS) {
      float s1 = 0.f, s2 = 0.f;
#pragma unroll
      for (int i = 0; i < 8; ++i) { float v = acc[i] + bv; s1 += v; s2 += v * v; }
      atomicAdd(&gsum[col], s1);
      atomicAdd(&gsum[C2_ + col], s2);
    } else {
      float sc = scsh[col], sh = scsh[C2_ + col];
#pragma unroll
      for (int i = 0; i < 8; ++i) {
        float v = fmaxf((acc[i] + bv) * sc + sh, 0.f);     // BN + ReLU
        int rm = rowTile + i + 8*hi;
        int bn2 = rm / KNN_;
        atomicMax(&gmax[(size_t)bn2 * C2_ + col], __float_as_uint(v)); // v>=0
      }
    }
  }
}

// ================= fc2: g(25088x192) * W2(192x96) + b2 =================
// A fragments register-resident, wave sweeps all 6 col tiles.
__global__ void __launch_bounds__(256) fc2_gemm(const float* __restrict__ g,
                                                const float* __restrict__ Wf,
                                                const float* __restrict__ bias,
                                                float* __restrict__ out) {
  __shared__ float Wl[C2_*16];

  int wave = threadIdx.x >> 5, lane = threadIdx.x & 31;
  int hi = lane >> 4, lp = lane & 15;
  int rowTile = (blockIdx.x * 8 + wave) * 16;
  const float* arow = g + (size_t)(rowTile + lp) * C2_;

  v2f afr[C2_/4];                                     // 96 VGPRs
#pragma unroll
  for (int k = 0; k < C2_/4; ++k) {
    int ka = 4*k + 2*hi;
    afr[k].x = arow[ka]; afr[k].y = arow[ka+1];
  }

  for (int ct = 0; ct < C_/16; ++ct) {
    int colBase = ct * 16;
    __syncthreads();
    for (int t = threadIdx.x; t < C2_*16; t += 256)
      Wl[t] = Wf[(t >> 4) * C_ + colBase + (t & 15)];
    __syncthreads();

    v8f acc = {};
#pragma unroll
    for (int k = 0; k < C2_/4; ++k) {
      int ka = 4*k + 2*hi;
      v2f bb; bb.x = Wl[ka*16 + lp]; bb.y = Wl[(ka+1)*16 + lp];
      acc = wmma4(afr[k], bb, acc);
    }
    float bv = bias[colBase + lp];
#pragma unroll
    for (int i = 0; i < 8; ++i)
      out[(size_t)(rowTile + i + 8*hi) * C_ + colBase + lp] = acc[i] + bv;
  }
}

// ================= bn2 apply + residual add, transpose to (B,C,N) =================
__global__ void add_shortcut(const float* __restrict__ t2, const float* __restrict__ scsh,
                             const float* __restrict__ x, float* __restrict__ z) {
  size_t i = (size_t)blockIdx.x * 256 + threadIdx.x;
  if (i < (size_t)B_ * C_ * N_) {
    int n = (int)(i % N_);
    int bc = (int)(i / N_);
    int c = bc % C_, b = bc / C_;
    size_t r = ((size_t)b * N_ + n) * C_ + c;
    z[i] = t2[r] * scsh[c] + scsh[C_ + c] + x[i];
  }
}

// ================= 3x3 stride-2 conv, 96->192, pad 1 =================
__global__ void conv3x3s2(const float* __restrict__ act, const float* __restrict__ Wd,
                          const float* __restrict__ bd, float* __restrict__ out) {
  int i = blockIdx.x * 256 + threadIdx.x;
  if (i >= B_ * OC_ * OHW_) return;
  int ox = i % 28; int t = i / 28;
  int oy = t % 28; t /= 28;
  int o = t % OC_; int b = t / OC_;
  float s = bd[o];
  const float* ab = act + (size_t)b * C_ * N_;
  const float* wb = Wd + (size_t)o * C_ * 9;
  int iy0 = oy * 2 - 1, ix0 = ox * 2 - 1;
  for (int c = 0; c < C_; ++c) {
    const float* ac = ab + (size_t)c * N_;
    const float* wc = wb + c * 9;
#pragma unroll
    for (int ky = 0; ky < 3; ++ky) {
      int iy = iy0 + ky;
      if ((unsigned)iy >= (unsigned)H_) continue;
#pragma unroll
      for (int kx = 0; kx < 3; ++kx) {
        int ix = ix0 + kx;
        if ((unsigned)ix >= (unsigned)W_) continue;
        s += ac[iy * W_ + ix] * wc[ky * 3 + kx];
      }
    }
  }
  out[i] = s;
}

// ======================================================================
extern "C" void kernel_launch(void* const* d_in, const int* in_sizes, int n_in,
                              void* d_out, int out_size, void* d_ws, size_t ws_size,
                              hipStream_t stream) {
  const float* x      = (const float*)d_in[0];
  const float* W_fc1  = (const float*)d_in[1];
  const float* b_fc1  = (const float*)d_in[2];
  const float* g_bn1  = (const float*)d_in[3];
  const float* be_bn1 = (const float*)d_in[4];
  const float* W_edge = (const float*)d_in[5];
  const float* b_edge = (const float*)d_in[6];
  const float* g_bne  = (const float*)d_in[7];
  const float* be_bne = (const float*)d_in[8];
  const float* W_fc2  = (const float*)d_in[9];
  const float* b_fc2  = (const float*)d_in[10];
  const float* g_bn2  = (const float*)d_in[11];
  const float* be_bn2 = (const float*)d_in[12];
  const float* g_sabn = (const float*)d_in[13];
  const float* be_sabn= (const float*)d_in[14];
  const float* W_dw   = (const float*)d_in[15];
  const float* b_dw   = (const float*)d_in[16];
  const float* g_dwbn = (const float*)d_in[17];
  const float* be_dwbn= (const float*)d_in[18];

  float* ws = (float*)d_ws;
  const size_t SZ_ROWxC  = (size_t)M1_ * C_;    // 2,408,448
  const size_t SZ_ROWxC2 = (size_t)M1_ * C2_;   // 4,816,896
  float* buf0   = ws;                                // yraw / t2
  float* buf1   = buf0 + SZ_ROWxC;                   // ft / z
  float* buf2   = buf1 + SZ_ROWxC;                   // g (uint-max) / act
  float* sqbuf  = buf2 + SZ_ROWxC2;
  int*   idxbuf = (int*)(sqbuf + M1_);
  float* convraw= (float*)(idxbuf + ME_);
  float* sums   = convraw + (size_t)B_ * OC_ * OHW_; // [2*192]
  float* scsh   = sums + 2 * C2_;                    // [2*192]

  const int T = 256;
  auto blks = [](size_t n) { return (unsigned)((n + 255) / 256); };

  // ---- fc1 + BN1 -> ft, |ft|^2 ----
  fc1_gemm<<<M1_/128, T, 0, stream>>>(x, W_fc1, b_fc1, buf0);
  stats_chlast<<<C_, T, 0, stream>>>(buf0, M1_, C_, sums);
  bn_finalize<<<1, C2_, 0, stream>>>(sums, g_bn1, be_bn1, C_, 1.f/(float)M1_, scsh);
  bn_apply_chlast<<<blks(SZ_ROWxC), T, 0, stream>>>(buf0, scsh, C_, SZ_ROWxC, buf1);
  rowsq<<<blks(M1_), T, 0, stream>>>(buf1, sqbuf);

  // ---- kNN top-9 via WMMA Gram tiles ----
  knn_topk<<<(B_*(N_/16))/8, T, 0, stream>>>(buf1, sqbuf, idxbuf);

  // ---- edge MLP pass 1: BN stats ----
  zero_f<<<blks(2*C2_), T, 0, stream>>>(sums, 2*C2_);
  edge_gemm<true><<<ME_/128, T, 0, stream>>>(buf1, idxbuf, W_edge, b_edge,
                                             nullptr, sums, nullptr);
  bn_finalize<<<1, C2_, 0, stream>>>(sums, g_bne, be_bne, C2_, 1.f/(float)ME_, scsh);

  // ---- edge MLP pass 2: BN+ReLU+max over K -> g ----
  zero_f<<<blks(SZ_ROWxC2), T, 0, stream>>>(buf2, SZ_ROWxC2);
  edge_gemm<false><<<ME_/128, T, 0, stream>>>(buf1, idxbuf, W_edge, b_edge,
                                              scsh, nullptr, (unsigned*)buf2);

  // ---- fc2 + BN2 + shortcut ----
  fc2_gemm<<<M1_/128, T, 0, stream>>>(buf2, W_fc2, b_fc2, buf0);
  stats_chlast<<<C_, T, 0, stream>>>(buf0, M1_, C_, sums);
  bn_finalize<<<1, C2_, 0, stream>>>(sums, g_bn2, be_bn2, C_, 1.f/(float)M1_, scsh);
  add_shortcut<<<blks(SZ_ROWxC), T, 0, stream>>>(buf0, scsh, x, buf1);   // z in (B,C,N)

  // ---- saBN + ReLU -> act ----
  stats_chmid<<<C_, T, 0, stream>>>(buf1, C_, N_, B_, sums);
  bn_finalize<<<1, C2_, 0, stream>>>(sums, g_sabn, be_sabn, C_, 1.f/(float)M1_, scsh);
  bn_relu_chmid<<<blks(SZ_ROWxC), T, 0, stream>>>(buf1, scsh, C_, N_, SZ_ROWxC, buf2);

  // ---- conv 3x3 s2 + dwBN + ReLU ----
  conv3x3s2<<<blks((size_t)B_*OC_*OHW_), T, 0, stream>>>(buf2, W_dw, b_dw, convraw);
  stats_chmid<<<OC_, T, 0, stream>>>(convraw, OC_, OHW_, B_, sums);
  bn_finalize<<<1, C2_, 0, stream>>>(sums, g_dwbn, be_dwbn, OC_, 1.f/(float)(B_*OHW_), scsh);
  bn_relu_chmid<<<blks((size_t)B_*OC_*OHW_), T, 0, stream>>>(convraw, scsh, OC_, OHW_,
                                                             (size_t)B_*OC_*OHW_, (float*)d_out);
}